// PBELoss_21869973471853
// MI455X (gfx1250) — compile-verified
//
#include <hip/hip_runtime.h>
#include <math.h>

// Problem constants (match the reference generator exactly).
#define NN 100000      // N_NODES
#define NE 3200000     // N_EDGES
#define NF 6           // N_FEAT

typedef float v2f __attribute__((ext_vector_type(2)));
typedef float v8f __attribute__((ext_vector_type(8)));

// ---------------------------------------------------------------------------
// Full wave32 sum using one V_WMMA_F32_16X16X4_F32.
//   A (16x4, 32-bit layout): lanes 0-15 hold {K0,K1}, lanes 16-31 hold {K2,K3}
//   We place lane partial p in K0/K2, zero in K1/K3. B = all ones (4x16).
//   => D[m][n] = p_m + p_{m+16} (identical for every column n).
//   D layout: lane n<16 holds rows 0..7 in d0..d7, lanes>=16 hold rows 8..15.
//   sum(d0..d7) + shfl_xor(.,16) == exact sum of all 32 lane partials.
// Requires EXEC == all ones (call from non-divergent code only).
// ---------------------------------------------------------------------------
__device__ __forceinline__ float wave_sum32(float p) {
#if __has_builtin(__builtin_amdgcn_wmma_f32_16x16x4_f32)
    v2f a;    a[0] = p;    a[1] = 0.0f;
    v2f ones; ones[0] = 1.0f; ones[1] = 1.0f;
    v8f c = {};
    v8f d = __builtin_amdgcn_wmma_f32_16x16x4_f32(
        /*neg_a=*/false, a, /*neg_b=*/false, ones,
        /*c_mod=*/(short)0, c, /*reuse_a=*/false, /*reuse_b=*/false);
    float s = d[0] + d[1] + d[2] + d[3] + d[4] + d[5] + d[6] + d[7];
    return s + __shfl_xor(s, 16, 32);
#else
    for (int off = 16; off > 0; off >>= 1) p += __shfl_xor(p, off, 32);
    return p;
#endif
}

// ---------------------------------------------------------------------------
// Kernel 1: per-node prep.
//   tp = where(mask, pred, target);  V = Vm * exp(j*Va)
//   Vr = Vm*cos(Va), Vi = Vm*sin(Va), netP = PG-PD, netQ = QG-QD
// ---------------------------------------------------------------------------
__global__ void node_prep_kernel(const float* __restrict__ pred,
                                 const float* __restrict__ target,
                                 const unsigned char* __restrict__ mask,
                                 float* __restrict__ Vr, float* __restrict__ Vi,
                                 float* __restrict__ netP, float* __restrict__ netQ) {
    int i = blockIdx.x * blockDim.x + threadIdx.x;
    if (i >= NN) return;
    const int b = i * NF;
    float tp[NF];
#pragma unroll
    for (int c = 0; c < NF; ++c)
        tp[c] = mask[b + c] ? pred[b + c] : target[b + c];
    float sn, cs;
    sincosf(tp[1], &sn, &cs);          // Va = col 1
    Vr[i]   = tp[0] * cs;              // Vm = col 0
    Vi[i]   = tp[0] * sn;
    netP[i] = tp[2] - tp[4];           // PG - PD
    netQ[i] = tp[3] - tp[5];           // QG - QD
}

// ---------------------------------------------------------------------------
// Kernel 2: edge scatter.  msg = (G - jB) * conj(V[col]); matvec[row] += msg.
// Streamed edge data loaded non-temporally so the L2 stays reserved for the
// node tables (Vr/Vi/mv ~1.6MB << 192MB L2 -> gathers & atomics are L2-local).
// ---------------------------------------------------------------------------
__global__ void edge_scatter_kernel(const int* __restrict__ src,
                                    const int* __restrict__ dst,
                                    const v2f* __restrict__ attr,   // (G,B) pairs
                                    const float* __restrict__ Vr,
                                    const float* __restrict__ Vi,
                                    float* __restrict__ mv_re,
                                    float* __restrict__ mv_im) {
    int e = blockIdx.x * blockDim.x + threadIdx.x;
    if (e >= NE) return;
    int row = __builtin_nontemporal_load(src + e);   // segment id
    int col = __builtin_nontemporal_load(dst + e);   // gather id
    v2f gb  = __builtin_nontemporal_load(attr + e);  // {G, B}
    float vr = Vr[col];
    float vi = Vi[col];
    // (G - jB)(vr - j vi) = (G*vr - B*vi) - j(G*vi + B*vr)
    float mr = fmaf(gb[0], vr, -gb[1] * vi);
    float mi = -fmaf(gb[0], vi,  gb[1] * vr);
    atomicAdd(&mv_re[row], mr);   // no-return global_atomic_add_f32
    atomicAdd(&mv_im[row], mi);
}

// ---------------------------------------------------------------------------
// Kernel 3: per-node loss partials + WMMA wave reduction + global atomics.
// No early return: every thread reaches wave_sum32 with EXEC all-1s.
// ---------------------------------------------------------------------------
__global__ void node_loss_kernel(const float* __restrict__ Vr,
                                 const float* __restrict__ Vi,
                                 const float* __restrict__ netP,
                                 const float* __restrict__ netQ,
                                 const float* __restrict__ mv_re,
                                 const float* __restrict__ mv_im,
                                 float* __restrict__ sums) {
    int i = blockIdx.x * blockDim.x + threadIdx.x;
    float pa = 0.0f, pr = 0.0f, pi = 0.0f;
    if (i < NN) {
        float vr = Vr[i], vi = Vi[i];
        float ar = mv_re[i], ai = mv_im[i];
        float Sre = fmaf(vr, ar, -vi * ai);   // S = V * matvec
        float Sim = fmaf(vr, ai,  vi * ar);
        float dr = netP[i] - Sre;
        float di = netQ[i] - Sim;
        pr = fabsf(dr);
        pi = fabsf(di);
        pa = sqrtf(fmaf(dr, dr, di * di));
    }
    float wa = wave_sum32(pa);
    float wr = wave_sum32(pr);
    float wi = wave_sum32(pi);
    if ((threadIdx.x & 31) == 0) {
        atomicAdd(&sums[0], wa);
        atomicAdd(&sums[1], wr);
        atomicAdd(&sums[2], wi);
    }
}

// Kernel 4: divide by N and emit the three scalars.
__global__ void finalize_kernel(const float* __restrict__ sums,
                                float* __restrict__ out) {
    if (blockIdx.x == 0 && threadIdx.x == 0) {
        const float inv = 1.0f / (float)NN;
        out[0] = sums[0] * inv;   // loss
        out[1] = sums[1] * inv;   // real_loss
        out[2] = sums[2] * inv;   // imag_loss
    }
}

extern "C" void kernel_launch(void* const* d_in, const int* in_sizes, int n_in,
                              void* d_out, int out_size, void* d_ws, size_t ws_size,
                              hipStream_t stream) {
    (void)in_sizes; (void)n_in; (void)out_size; (void)ws_size;

    const float*         pred      = (const float*)d_in[0];
    const float*         target    = (const float*)d_in[1];
    const int*           edge_idx  = (const int*)d_in[2];        // (2, NE)
    const float*         edge_attr = (const float*)d_in[3];      // (NE, 2)
    const unsigned char* mask      = (const unsigned char*)d_in[4]; // bool (NN, 6)

    const int* src = edge_idx;        // row (segment ids)
    const int* dst = edge_idx + NE;   // col (gather ids)

    // Workspace layout (floats): [mv_re NN][mv_im NN][Vr NN][Vi NN][netP NN][netQ NN][sums 3]
    float* ws    = (float*)d_ws;
    float* mv_re = ws;
    float* mv_im = ws + NN;
    float* Vr    = ws + 2 * NN;
    float* Vi    = ws + 3 * NN;
    float* netP  = ws + 4 * NN;
    float* netQ  = ws + 5 * NN;
    float* sums  = ws + 6 * NN;

    // Zero the accumulators every call (graph-capture-safe memsets).
    hipMemsetAsync(mv_re, 0, 2 * (size_t)NN * sizeof(float), stream);
    hipMemsetAsync(sums,  0, 3 * sizeof(float), stream);

    const int BLK = 256;
    node_prep_kernel<<<(NN + BLK - 1) / BLK, BLK, 0, stream>>>(
        pred, target, mask, Vr, Vi, netP, netQ);

    edge_scatter_kernel<<<(NE + BLK - 1) / BLK, BLK, 0, stream>>>(
        src, dst, (const v2f*)edge_attr, Vr, Vi, mv_re, mv_im);

    node_loss_kernel<<<(NN + BLK - 1) / BLK, BLK, 0, stream>>>(
        Vr, Vi, netP, netQ, mv_re, mv_im, sums);

    finalize_kernel<<<1, 32, 0, stream>>>(sums, (float*)d_out);
}